// VNDGCNN3D_51032801411177
// MI455X (gfx1250) — compile-verified
//
#include <hip/hip_runtime.h>

// ---------------------------------------------------------------------------
// VN-DGCNN fused pipeline for MI455X (gfx1250, wave32, WMMA f32 16x16x4)
//   k1: per-point top-20 neighbor selection (8192 rows of 256 dists)
//   k2: fused gather + layer1(64x3) + layer2(128x64 WMMA) + layer3(256x128
//       WMMA) + max over k=20  -> hmax (B,256,N) in workspace
//   k3: fc1  G = relu(hmax @ fc1^T + b1)   (M=256,N=128,K=256 per batch, WMMA)
//   k4: fc2  out = G @ fc2^T + b2          (M=256,N=40 ,K=128 per batch, WMMA)
//
// H1/H2 live in LDS in a K-pair-interleaved layout: element (k, col) is at
// pair[(k>>1)*COLS + col].{x,y} so every WMMA B-fragment (which needs K=krow
// and K=krow+1 with krow even) is ONE aligned ds_load_b64 into an even VGPR
// pair -- no repacking movs.
// ---------------------------------------------------------------------------

typedef __attribute__((ext_vector_type(2))) float v2f;
typedef __attribute__((ext_vector_type(8))) float v8f;

__device__ __forceinline__ v8f wmma4(v2f a, v2f b, v8f c) {
  // D = A(16x4) * B(4x16) + C(16x16), fp32 throughout
  return __builtin_amdgcn_wmma_f32_16x16x4_f32(
      /*neg_a=*/false, a, /*neg_b=*/false, b,
      /*c_mod=*/(short)0, c, /*reuse_a=*/false, /*reuse_b=*/false);
}

#define BATCH 32
#define NPTS  256
#define KNB   20
#define PPW   4              // positions per workgroup
#define COLS  (PPW * KNB)    // 80 columns per workgroup

// ---------------------------------------------------------------------------
// Kernel 1: top-20 nearest neighbors per point. One block per batch, one
// thread per point; x[b] staged in LDS; fully-unrolled insertion keeps the
// 20-entry sorted list in VGPRs. The insertion chain is skipped for the whole
// wave when no lane can enter the list (v <= best[19] == list minimum), which
// is the common case late in the scan.
// Tie-break: strict '>' keeps the earlier index first, matching lax.top_k.
// ---------------------------------------------------------------------------
__global__ __launch_bounds__(256) void topk_kernel(const float* __restrict__ x,
                                                   int* __restrict__ idx_out) {
  __shared__ float sx[NPTS * 3];
  const int b = blockIdx.x, tid = threadIdx.x;
  for (int e = tid; e < NPTS * 3; e += 256) sx[e] = x[b * NPTS * 3 + e];
  __syncthreads();

  const float px = sx[tid * 3 + 0];
  const float py = sx[tid * 3 + 1];
  const float pz = sx[tid * 3 + 2];

  float best[KNB];
  int bidx[KNB];
#pragma unroll
  for (int j = 0; j < KNB; ++j) { best[j] = -__builtin_huge_valf(); bidx[j] = 0; }

  for (int m = 0; m < NPTS; ++m) {
    const float dx = px - sx[m * 3 + 0];
    const float dy = py - sx[m * 3 + 1];
    const float dz = pz - sx[m * 3 + 2];
    float v = -(dx * dx + dy * dy + dz * dz);  // -||xi-xj||^2 (self -> 0, max)
    if (__any(v > best[KNB - 1])) {            // wave-uniform early-out
      int id = m;
#pragma unroll
      for (int j = 0; j < KNB; ++j) {
        if (v > best[j]) {
          float tv = best[j]; best[j] = v; v = tv;
          int   ti = bidx[j]; bidx[j] = id; id = ti;
        }
      }
    }
  }
#pragma unroll
  for (int j = 0; j < KNB; ++j)
    idx_out[(b * NPTS + tid) * KNB + j] = bidx[j];
}

// ---------------------------------------------------------------------------
// Kernel 2: fused VN layers + max over k.
// Per workgroup: 4 positions x 20 neighbors = 80 columns, 256 threads = 8 waves.
// LDS: coords (80x3), H1 (64x80, pair-interleaved), H2 (128x80, pair-
// interleaved). H1's first 4 KB is recycled as hmax[256][4] after layer2.
// Layer2: wave wv owns Mtile wv; w2 A-fragments cached in VGPRs; 5 N-tiles x
// 16 K-steps. Layer3: Mtiles {wv, wv+8}; 5 N-tiles x 32 K-steps. Max over k
// folded in with ds int-atomic-max (relu outputs >= 0 -> int order is exact).
// ---------------------------------------------------------------------------
__global__ __launch_bounds__(256) void vn_fused_kernel(
    const float* __restrict__ x, const int* __restrict__ nbr,
    const float* __restrict__ w1, const float* __restrict__ s1, const float* __restrict__ t1,
    const float* __restrict__ w2, const float* __restrict__ s2, const float* __restrict__ t2,
    const float* __restrict__ w3, const float* __restrict__ s3, const float* __restrict__ t3,
    float* __restrict__ hmax_out) {
  __shared__ float sNb[COLS * 3];
  __shared__ float sH1[64 * COLS];    // pair layout; head recycled as hmax
  __shared__ float sH2[128 * COLS];   // pair layout
  v2f* const sH1p = (v2f*)sH1;
  v2f* const sH2p = (v2f*)sH2;

  const int tid  = threadIdx.x;
  const int wv   = tid >> 5;
  const int lane = tid & 31;
  const int half = lane >> 4;   // 0: lanes 0-15, 1: lanes 16-31
  const int lm   = lane & 15;
  const int pos0 = blockIdx.x * PPW;

  // Pull the (L2-resident) weight matrices toward this WGP before we need them.
  __builtin_prefetch(((const char*)w2) + tid * 128, 0, 0);   // 32 KB
  __builtin_prefetch(((const char*)w3) + tid * 512, 0, 0);   // 128 KB

  // ---- gather neighbor coordinates --------------------------------------
  if (tid < COLS) {
    const int p   = tid / KNB, j = tid % KNB;
    const int pos = pos0 + p;
    const int b   = pos >> 8;
    const int nb  = nbr[pos * KNB + j];
    const float* xp = x + (b * NPTS + nb) * 3;
    sNb[tid * 3 + 0] = xp[0];
    sNb[tid * 3 + 1] = xp[1];
    sNb[tid * 3 + 2] = xp[2];
  }
  __syncthreads();

  // ---- layer1: H1[c][col] = relu(s1*(w1 . xyz) + t1), channel pairs -----
  for (int e = tid; e < 32 * COLS; e += 256) {
    const int c2 = e / COLS, col = e % COLS;
    const int c = c2 * 2;
    const float nx = sNb[col * 3], ny = sNb[col * 3 + 1], nz = sNb[col * 3 + 2];
    float v0 = w1[c * 3 + 0] * nx + w1[c * 3 + 1] * ny + w1[c * 3 + 2] * nz;
    float v1 = w1[c * 3 + 3] * nx + w1[c * 3 + 4] * ny + w1[c * 3 + 5] * nz;
    v2f pr;
    pr.x = fmaxf(v0 * s1[c] + t1[c], 0.0f);
    pr.y = fmaxf(v1 * s1[c + 1] + t1[c + 1], 0.0f);
    sH1p[c2 * COLS + col] = pr;           // ds_store_b64
  }
  __syncthreads();

  // ---- layer2: H2 = relu(s2 * (W2[128x64] @ H1[64x80]) + t2) ------------
  {
    v2f afrag[16];
#pragma unroll
    for (int kk = 0; kk < 16; ++kk) {
      const float* wr = w2 + (wv * 16 + lm) * 64 + kk * 4 + 2 * half;
      afrag[kk] = *(const v2f*)wr;        // global_load_b64
    }
    float sc[8], tc[8];
#pragma unroll
    for (int r = 0; r < 8; ++r) {
      const int c = wv * 16 + r + 8 * half;
      sc[r] = s2[c]; tc[r] = t2[c];
    }
#pragma unroll
    for (int nt = 0; nt < 5; ++nt) {
      const int col = nt * 16 + lm;
      v8f acc = {};
#pragma unroll
      for (int kk = 0; kk < 16; ++kk) {
        const v2f bf = sH1p[(kk * 2 + half) * COLS + col];   // ds_load_b64
        acc = wmma4(afrag[kk], bf, acc);
      }
#pragma unroll
      for (int r = 0; r < 8; r += 2) {
        const int c0 = wv * 16 + r + 8 * half;               // even
        v2f pr;
        pr.x = fmaxf(acc[r]     * sc[r]     + tc[r],     0.0f);
        pr.y = fmaxf(acc[r + 1] * sc[r + 1] + tc[r + 1], 0.0f);
        sH2p[(c0 >> 1) * COLS + col] = pr;                   // ds_store_b64
      }
    }
  }
  __syncthreads();

  // ---- recycle H1's head as hmax[256][PPW], zero it ----------------------
  float* const hmax = sH1;
  for (int e = tid; e < 256 * PPW; e += 256) hmax[e] = 0.0f;
  __syncthreads();

  // ---- layer3 + max over k: relu(s3*(W3[256x128] @ H2[128x80]) + t3) ----
  for (int mh = 0; mh < 2; ++mh) {
    const int mt = wv + 8 * mh;
    v2f afrag[32];
#pragma unroll
    for (int kk = 0; kk < 32; ++kk) {
      const float* wr = w3 + (mt * 16 + lm) * 128 + kk * 4 + 2 * half;
      afrag[kk] = *(const v2f*)wr;        // global_load_b64
    }
    float sc[8], tc[8];
#pragma unroll
    for (int r = 0; r < 8; ++r) {
      const int c = mt * 16 + r + 8 * half;
      sc[r] = s3[c]; tc[r] = t3[c];
    }
#pragma unroll
    for (int nt = 0; nt < 5; ++nt) {
      const int col = nt * 16 + lm;
      v8f acc = {};
#pragma unroll
      for (int kk = 0; kk < 32; ++kk) {
        const v2f bf = sH2p[(kk * 2 + half) * COLS + col];   // ds_load_b64
        acc = wmma4(afrag[kk], bf, acc);
      }
      const int p = col / KNB;
#pragma unroll
      for (int r = 0; r < 8; ++r) {
        const int c = mt * 16 + r + 8 * half;
        const float v = fmaxf(acc[r] * sc[r] + tc[r], 0.0f);
        atomicMax((int*)&hmax[c * PPW + p], __float_as_int(v));  // ds_max_i32
      }
    }
  }
  __syncthreads();

  // ---- write hmax to global (B, 256, N) ---------------------------------
  for (int e = tid; e < 256 * PPW; e += 256) {
    const int c = e / PPW, p = e % PPW;
    const int pos = pos0 + p;
    const int b = pos >> 8, n = pos & 255;
    hmax_out[(b * 256 + c) * NPTS + n] = hmax[e];
  }
}

// ---------------------------------------------------------------------------
// Kernel 3: fc1 per batch: G[b] = relu(hmax[b] @ fc1^T + b1)
// M=256 (channels), N=128 (fc1 out), K=256 (points). One block per batch,
// 8 waves; wave wv owns N-tile wv and loops 16 M-tiles x 64 K-steps of WMMA.
// A/B fragments are contiguous aligned 8-byte loads -> global_load_b64.
// ---------------------------------------------------------------------------
__global__ __launch_bounds__(256) void fc1_kernel(const float* __restrict__ hm,
                                                  const float* __restrict__ fw,
                                                  const float* __restrict__ fb,
                                                  float* __restrict__ g) {
  const int b = blockIdx.x;
  const int tid = threadIdx.x, wv = tid >> 5, lane = tid & 31;
  const int half = lane >> 4, lm = lane & 15;
  const int o = wv * 16 + lm;                 // output feature (N dim)
  const float* A0 = hm + b * 256 * NPTS;
  const v2f* brp = (const v2f*)(fw + o * 256 + 2 * half);  // B[k][o]=fc1_w[o*256+k]
  const float bias = fb[o];

  for (int mt = 0; mt < 16; ++mt) {
    const v2f* arp = (const v2f*)(A0 + (mt * 16 + lm) * NPTS + 2 * half);
    v8f acc = {};
#pragma unroll 8
    for (int kk = 0; kk < 64; ++kk) {
      acc = wmma4(arp[kk * 2], brp[kk * 2], acc);
    }
#pragma unroll
    for (int r = 0; r < 8; ++r) {
      const int c = mt * 16 + r + 8 * half;
      g[(b * 256 + c) * 128 + o] = fmaxf(acc[r] + bias, 0.0f);
    }
  }
}

// ---------------------------------------------------------------------------
// Kernel 4: fc2 per batch: out[b] = G[b] @ fc2^T + b2
// M=256, N=40 (padded to 48, guarded), K=128. 48 tiles over 8 waves.
// ---------------------------------------------------------------------------
__global__ __launch_bounds__(256) void fc2_kernel(const float* __restrict__ g,
                                                  const float* __restrict__ fw,
                                                  const float* __restrict__ fb,
                                                  float* __restrict__ out) {
  const int b = blockIdx.x;
  const int tid = threadIdx.x, wv = tid >> 5, lane = tid & 31;
  const int half = lane >> 4, lm = lane & 15;
  const v2f zero2 = {0.0f, 0.0f};

  for (int i = 0; i < 6; ++i) {
    const int t = wv + 8 * i;       // 0..47
    const int mt = t / 3, nt = t % 3;
    const int q = nt * 16 + lm;
    const bool qv = (q < 40);
    const v2f* arp = (const v2f*)(g + (b * 256 + mt * 16 + lm) * 128 + 2 * half);
    const v2f* brp = (const v2f*)(fw + (qv ? q : 0) * 128 + 2 * half);
    v8f acc = {};
#pragma unroll 8
    for (int kk = 0; kk < 32; ++kk) {
      const v2f bf = qv ? brp[kk * 2] : zero2;
      acc = wmma4(arp[kk * 2], bf, acc);
    }
    if (qv) {
      const float bias = fb[q];
#pragma unroll
      for (int r = 0; r < 8; ++r) {
        const int c = mt * 16 + r + 8 * half;
        out[(b * 256 + c) * 40 + q] = acc[r] + bias;
      }
    }
  }
}

// ---------------------------------------------------------------------------
extern "C" void kernel_launch(void* const* d_in, const int* in_sizes, int n_in,
                              void* d_out, int out_size, void* d_ws, size_t ws_size,
                              hipStream_t stream) {
  const float* x     = (const float*)d_in[0];
  const float* w1    = (const float*)d_in[1];
  const float* s1    = (const float*)d_in[2];
  const float* t1    = (const float*)d_in[3];
  const float* w2    = (const float*)d_in[4];
  const float* s2    = (const float*)d_in[5];
  const float* t2    = (const float*)d_in[6];
  const float* w3    = (const float*)d_in[7];
  const float* s3    = (const float*)d_in[8];
  const float* t3    = (const float*)d_in[9];
  const float* fc1_w = (const float*)d_in[10];
  const float* fc1_b = (const float*)d_in[11];
  const float* fc2_w = (const float*)d_in[12];
  const float* fc2_b = (const float*)d_in[13];
  float* outp = (float*)d_out;

  // Workspace layout
  const size_t idx_bytes  = (size_t)BATCH * NPTS * KNB * sizeof(int);    // 640 KB
  const size_t hmax_bytes = (size_t)BATCH * 256 * NPTS * sizeof(float);  // 8 MB
  int*   ws_idx  = (int*)d_ws;
  float* ws_hmax = (float*)((char*)d_ws + idx_bytes);
  float* ws_g    = (float*)((char*)d_ws + idx_bytes + hmax_bytes);       // 4 MB

  topk_kernel<<<BATCH, 256, 0, stream>>>(x, ws_idx);

  const int n_wg = (BATCH * NPTS) / PPW;  // 2048
  vn_fused_kernel<<<n_wg, 256, 0, stream>>>(x, ws_idx,
                                            w1, s1, t1,
                                            w2, s2, t2,
                                            w3, s3, t3,
                                            ws_hmax);

  fc1_kernel<<<BATCH, 256, 0, stream>>>(ws_hmax, fc1_w, fc1_b, ws_g);
  fc2_kernel<<<BATCH, 256, 0, stream>>>(ws_g, fc2_w, fc2_b, outp);
}